// AllPassMORRCirculantLinear_38517266711180
// MI455X (gfx1250) — compile-verified
//
#include <hip/hip_runtime.h>
#include <hip/hip_bf16.h>

// CDNA5 / gfx1250, wave32.
typedef __attribute__((ext_vector_type(2))) float v2f;
typedef __attribute__((ext_vector_type(8))) float v8f;

// Constants: A=0.8682, R=0.8602
// S = A^2+R^2, K1 = 2AR, D = 1+(AR)^2
#define MRR_S  1.49371528f
#define MRR_K1 1.49365128f
#define MRR_D  1.55774850f
#define INV_2PI 0.15915494309189535f

__device__ __forceinline__ float mrr_tr(float phi) {
    // v_cos_f32 computes cos(2*pi*x) -> pre-scale by 1/(2*pi)
    float cv  = __builtin_amdgcn_cosf(phi * INV_2PI);
    float c   = MRR_K1 * cv;
    float num = MRR_S - c;
    float den = MRR_D - c;
    return num * __builtin_amdgcn_rcpf(den);
}

// Build scale[0..127] from morr_output_scale (65 values):
// s = m - mean(m); scale[q<64] = s[q]; scale[q>=64] = -s[q-64]
__global__ void morr_scale_kernel(const float* __restrict__ m,
                                  float* __restrict__ scale) {
    int tid = threadIdx.x; // 128 threads
    float sum = 0.0f;
    #pragma unroll
    for (int i = 0; i < 65; ++i) sum += m[i];
    float mean = sum * (1.0f / 65.0f);
    float v = (tid < 64) ? (m[tid] - mean) : -(m[tid - 64] - mean);
    scale[tid] = v;
}

// One wave computes a 16x16 tile of Out[1024 x 1024] (rows = batch b,
// cols = p*8 + t). For each q: two V_WMMA_F32_16X16X4_F32 chained through C
// produce phi (K=8 circular conv), then fused nonlinearity + scaled reduce.
// Block = 256 threads = 8 waves; all waves share one p-pair -> |w| staged in LDS.
// grid = (64 col-tiles, 8 row-groups)
__global__ void __launch_bounds__(256)
morr_circulant_kernel(const float* __restrict__ x,       // (1024, 1024)
                      const float* __restrict__ weight,  // (128, 128, 8)
                      const float* __restrict__ scale_g, // (128,) from d_ws
                      float* __restrict__ out) {         // (1024, 1024)
    __shared__ float wlds[2048]; // |w[p0:p0+2, 0:128, 0:8]|
    __shared__ float slds[128];  // scale[q]

    const int tid = threadIdx.x;
    const int p0base = blockIdx.x * 2048; // p0 = 2*blockIdx.x, rows of 1024

    // Stage |weight| for this block's two p values (contiguous 2048 floats)
    #pragma unroll
    for (int i = tid; i < 2048; i += 256)
        wlds[i] = fabsf(weight[p0base + i]);
    if (tid < 128) slds[tid] = scale_g[tid];
    __syncthreads();

    const int warp = tid >> 5;
    const int lane = tid & 31;
    const int half = lane >> 4;   // 0: lanes 0-15, 1: lanes 16-31
    const int mn   = lane & 15;   // M index (A) / N index (B,C,D)
    const int kb   = half * 2;    // K base held by this half-wave (V0,V1 = kb,kb+1)

    const int b0 = (blockIdx.y * 8 + warp) * 16; // row tile base (batch)
    const int c0 = blockIdx.x * 16;              // col tile base (p*8+t)

    // B-operand circulant gather offsets into wlds (q term added per iter):
    // Wt[s, n] = |w[p_local, q, (t - s) & 7]|,  n = p_local*8 + t
    const int pl = mn >> 3;
    const int tt = mn & 7;
    const int boff0 = pl * 1024 + ((tt - kb)     & 7);
    const int boff1 = pl * 1024 + ((tt - kb - 1) & 7);
    const int boff2 = pl * 1024 + ((tt - kb - 4) & 7);
    const int boff3 = pl * 1024 + ((tt - kb - 5) & 7);

    const float* xrow = x + (b0 + mn) * 1024;

    v8f acc = {};

    for (int q = 0; q < 128; ++q) {
        // A operand: xb = x^2, 16x8 tile split as two 16x4 operands.
        const int xo = q * 8 + kb;               // even -> 8B aligned
        const float2 xa = *(const float2*)(xrow + xo);
        const float2 xc = *(const float2*)(xrow + xo + 4);
        v2f a_lo = { xa.x * xa.x, xa.y * xa.y };
        v2f a_hi = { xc.x * xc.x, xc.y * xc.y };

        // B operand from LDS (circulant of |w|)
        const int qb = q * 8;
        v2f b_lo = { wlds[qb + boff0], wlds[qb + boff1] };
        v2f b_hi = { wlds[qb + boff2], wlds[qb + boff3] };

        // phi = A(:,0:4) @ B(0:4,:) + A(:,4:8) @ B(4:8,:)
        v8f phi = {};
        phi = __builtin_amdgcn_wmma_f32_16x16x4_f32(
            false, a_lo, false, b_lo, (short)0, phi, false, false);
        phi = __builtin_amdgcn_wmma_f32_16x16x4_f32(
            false, a_hi, false, b_hi, (short)0, phi, false, false);

        // Fused nonlinearity + scaled accumulation over q
        const float sc = slds[q];
        #pragma unroll
        for (int i = 0; i < 8; ++i)
            acc[i] = fmaf(sc, mrr_tr(phi[i]), acc[i]);
    }

    // D layout: VGPR i holds row M = half*8 + i, col N = mn
    const int mbase = half * 8;
    #pragma unroll
    for (int i = 0; i < 8; ++i)
        out[(size_t)(b0 + mbase + i) * 1024 + c0 + mn] = acc[i];
}

extern "C" void kernel_launch(void* const* d_in, const int* in_sizes, int n_in,
                              void* d_out, int out_size, void* d_ws, size_t ws_size,
                              hipStream_t stream) {
    const float* x      = (const float*)d_in[0]; // (1024,1024)
    const float* weight = (const float*)d_in[1]; // (128,128,8)
    const float* morr   = (const float*)d_in[2]; // (65,)
    float* out   = (float*)d_out;                // (1024,1024)
    float* scale = (float*)d_ws;                 // 128 floats scratch

    morr_scale_kernel<<<1, 128, 0, stream>>>(morr, scale);

    dim3 grid(64, 8); // 64 col tiles x (8 row-groups * 8 waves) = full 1024x1024
    morr_circulant_kernel<<<grid, 256, 0, stream>>>(x, weight, scale, out);
}